// LocalConv_37271726195395
// MI455X (gfx1250) — compile-verified
//
#include <hip/hip_runtime.h>

// ---------------------------------------------------------------------------
// Locally-connected conv (B=64, 32x32, CIN=COUT=64, 3x3 SAME) for gfx1250.
// One workgroup per output pixel -> GEMM  Y[64x64] = Patch[64x576] * W[576x64].
// 4 wave32s per block; each wave owns a 32x32 quadrant = 2x2 grid of 16x16
// tiles (register blocking: 4 WMMAs per 4 ds_load_b64), K in 3 chunks of 192
// staged through LDS, computed with V_WMMA_F32_16X16X4_F32.
// ---------------------------------------------------------------------------

typedef __attribute__((ext_vector_type(2))) float v2f;
typedef __attribute__((ext_vector_type(4))) float v4f;
typedef __attribute__((ext_vector_type(8))) float v8f;

#define BATCH    64
#define HH       32
#define WW       32
#define CIN      64
#define COUT     64
#define KCHUNK   192            // one kh row: 3 * CIN
#define LDSSTR   196            // KCHUNK + 4 pad -> bank stride 4, conflict-free
#define NTHREADS 128            // 4 wave32s

#define WMMA_F32(A, B, C)                                            \
  __builtin_amdgcn_wmma_f32_16x16x4_f32(false, (A), false, (B),      \
                                        (short)0, (C), false, false)

__global__ __launch_bounds__(NTHREADS)
void localconv_wmma_f32(const float* __restrict__ x,
                        const float* __restrict__ ker,
                        const float* __restrict__ bias,
                        float* __restrict__ out) {
  extern __shared__ float lds[];
  float* As = lds;                      // [BATCH][LDSSTR]  patch rows (M x K)
  float* Bs = lds + BATCH * LDSSTR;     // [COUT ][LDSSTR]  weights transposed (N x K)

  const int h    = blockIdx.x >> 5;
  const int w    = blockIdx.x & 31;
  const int t    = threadIdx.x;
  const int lane = t & 31;
  const int wv   = t >> 5;              // wave id 0..3
  const int m_base = (wv >> 1) << 5;    // 0 or 32
  const int n_base = (wv & 1) << 5;     // 0 or 32
  const int lh = lane >> 4;             // half-wave select (K offset +2)
  const int ll = lane & 15;

  // per-location unshared weights: ker[h][w][k=0..575][co=0..63]
  const float* ker_loc = ker + (size_t)((h * WW + w) * (9 * CIN)) * COUT;

  v8f c00 = {}, c01 = {}, c10 = {}, c11 = {};   // 2x2 tile accumulators

  for (int kh = 0; kh < 3; ++kh) {
    __syncthreads();
    const int y = h + kh - 1;

    // ---- stage A chunk: As[b][kw*64+ci] = x[b, y, w+kw-1, ci], zero-padded.
    // 64 rows x 192 cols as float4: 3072 vec4 / 128 threads = 24 iters.
    for (int idx = t; idx < BATCH * (KCHUNK / 4); idx += NTHREADS) {
      const int b  = idx / (KCHUNK / 4);
      const int kk = (idx % (KCHUNK / 4)) * 4;
      const int kw = kk >> 6;                       // 0..2
      const int ci = kk & 63;
      const int xw = w + kw - 1;
      v4f v = {};
      if ((unsigned)y < HH && (unsigned)xw < WW) {
        v = *(const v4f*)(x + (size_t)(((b * HH + y) * WW + xw) * CIN + ci));
      }
      *(v4f*)(As + b * LDSSTR + kk) = v;
    }

    // ---- stage B chunk transposed: Bs[co][kk] = ker_loc[(kh*192+kk)*64+co].
    // Coalesced non-temporal reads along co (weights are single-use: stream).
    for (int idx = t; idx < KCHUNK * (COUT / 4); idx += NTHREADS) {
      const int kk  = idx / (COUT / 4);
      const int co4 = (idx % (COUT / 4)) * 4;
      const v4f v = __builtin_nontemporal_load(
          (const v4f*)(ker_loc + (size_t)((kh * KCHUNK + kk) * COUT + co4)));
      Bs[(co4 + 0) * LDSSTR + kk] = v.x;
      Bs[(co4 + 1) * LDSSTR + kk] = v.y;
      Bs[(co4 + 2) * LDSSTR + kk] = v.z;
      Bs[(co4 + 3) * LDSSTR + kk] = v.w;
    }

    // ---- prefetch next weight chunk into L2 while computing this one.
    // 384 cachelines x 128B = 48KB = exactly one chunk; 3 per thread.
    if (kh < 2) {
      const float* nxt = ker_loc + (size_t)((kh + 1) * KCHUNK) * COUT;
      for (int i = t; i < (KCHUNK * COUT) / 32; i += NTHREADS) {
        __builtin_prefetch(nxt + i * 32, 0, 1);
      }
    }
    __syncthreads();

    // ---- 48 K-steps x 4 WMMAs over this chunk (1 ds_load_b64 per WMMA).
    // A 16x4 layout: lanes 0-15 M=0..15 {VGPR0=K, VGPR1=K+1}; lanes 16-31 K+2/3
    // B 4x16 layout: lane ll = N, VGPRs = K rows; Bs is N-major -> same b64 shape
    const float* a0p = As + (m_base + ll) * LDSSTR + 2 * lh;
    const float* a1p = a0p + 16 * LDSSTR;
    const float* b0p = Bs + (n_base + ll) * LDSSTR + 2 * lh;
    const float* b1p = b0p + 16 * LDSSTR;
#pragma unroll
    for (int kk = 0; kk < KCHUNK; kk += 4) {
      const v2f a0 = *(const v2f*)(a0p + kk);
      const v2f a1 = *(const v2f*)(a1p + kk);
      const v2f b0 = *(const v2f*)(b0p + kk);
      const v2f b1 = *(const v2f*)(b1p + kk);
      c00 = WMMA_F32(a0, b0, c00);      // 4 independent accumulators:
      c01 = WMMA_F32(a0, b1, c01);      // no WMMA->WMMA RAW hazards
      c10 = WMMA_F32(a1, b0, c10);
      c11 = WMMA_F32(a1, b1, c11);
    }
  }

  // ---- epilogue: add bias, store. C/D layout: VGPR g, lane: M = g + 8*lh,
  // N = ll  ->  y[b, h, w, co]
  const float* bias_loc = bias + (h * WW + w) * COUT;
  const float bv0 = bias_loc[n_base + ll];
  const float bv1 = bias_loc[n_base + 16 + ll];
#pragma unroll
  for (int g = 0; g < 8; ++g) {
    const int m0 = m_base + g + 8 * lh;
    const int m1 = m0 + 16;
    float* r0 = out + (size_t)((m0 * HH + h) * WW + w) * COUT;
    float* r1 = out + (size_t)((m1 * HH + h) * WW + w) * COUT;
    r0[n_base + ll]      = c00[g] + bv0;
    r0[n_base + 16 + ll] = c01[g] + bv1;
    r1[n_base + ll]      = c10[g] + bv0;
    r1[n_base + 16 + ll] = c11[g] + bv1;
  }
}

extern "C" void kernel_launch(void* const* d_in, const int* in_sizes, int n_in,
                              void* d_out, int out_size, void* d_ws, size_t ws_size,
                              hipStream_t stream) {
  const float* x    = (const float*)d_in[0];   // [64,32,32,64]
  const float* ker  = (const float*)d_in[1];   // [32,32,576,64]
  const float* bias = (const float*)d_in[2];   // [32,32,64]
  float*       out  = (float*)d_out;           // [64,32,32,64]

  const dim3 grid(HH * WW);                    // one block per output pixel
  const dim3 block(NTHREADS);
  const size_t shmem = (size_t)2 * BATCH * LDSSTR * sizeof(float);  // ~100 KB

  hipLaunchKernelGGL(localconv_wmma_f32, grid, block, shmem, stream,
                     x, ker, bias, out);
}